// LstmModel_11287174054288
// MI455X (gfx1250) — compile-verified
//
#include <hip/hip_runtime.h>
#include <hip/hip_bf16.h>

// LSTM: SEQ=65536, IN=64, H=128 (4H=512 gates)
// Phase 1: x_gates[SEQ,512] = input @ W_ih.T + b_ih + b_hh  (fp32 WMMA GEMM, mem-bound)
// Phase 2: sequential scan, single workgroup, W_hh resident in VGPRs, v_pk_fma_f32 matvec
// Phase 3: sigmoid(h_T @ W_lin.T + b_lin) -> d_out[0]

#define SEQ 65536
#define IN  64
#define HD  128
#define G4  512   // 4*H

typedef float v2f __attribute__((ext_vector_type(2)));
typedef float v8f __attribute__((ext_vector_type(8)));

// ---------------------------------------------------------------------------
// Phase 1: GEMM via V_WMMA_F32_16X16X4_F32.
// Grid: SEQ/16 blocks of 256 threads (8 waves). Wave w covers columns
// [64*w, 64*w+64) of the 512 gate columns for this block's 16 rows.
// A-frag (16x4 f32): lane l<16 holds A[l][k],A[l][k+1]; lane>=16 holds k+2,k+3.
// B-frag (4x16 f32): lane l<16 holds B[k][l],B[k+1][l] == Wih[col][k..k+1].
// C/D (16x16 f32, 8 VGPRs): reg r -> row m0+r+8*(lane>=16), col n0+(lane&15).
// ---------------------------------------------------------------------------
__global__ __launch_bounds__(256) void xgates_gemm(
    const float* __restrict__ x,     // [SEQ, IN]
    const float* __restrict__ Wih,   // [G4, IN]
    const float* __restrict__ bih,   // [G4]
    const float* __restrict__ bhh,   // [G4]
    float* __restrict__ xg)          // [SEQ, G4]
{
  const int lane = threadIdx.x & 31;
  const int wave = threadIdx.x >> 5;        // 0..7
  const int m0   = blockIdx.x * 16;
  const int n0   = wave * 64;
  const int half = lane >> 4;               // 0 -> k,k+1 ; 1 -> k+2,k+3
  const int l16  = lane & 15;

  v8f acc0 = {}, acc1 = {}, acc2 = {}, acc3 = {};

  const float* arow = x + (size_t)(m0 + l16) * IN + 2 * half;
  #pragma unroll
  for (int k = 0; k < IN; k += 4) {
    v2f a = *(const v2f*)(arow + k);
    v2f b0 = *(const v2f*)(Wih + (size_t)(n0 +  0 + l16) * IN + k + 2 * half);
    v2f b1 = *(const v2f*)(Wih + (size_t)(n0 + 16 + l16) * IN + k + 2 * half);
    v2f b2 = *(const v2f*)(Wih + (size_t)(n0 + 32 + l16) * IN + k + 2 * half);
    v2f b3 = *(const v2f*)(Wih + (size_t)(n0 + 48 + l16) * IN + k + 2 * half);
    acc0 = __builtin_amdgcn_wmma_f32_16x16x4_f32(false, a, false, b0, (short)0, acc0, false, false);
    acc1 = __builtin_amdgcn_wmma_f32_16x16x4_f32(false, a, false, b1, (short)0, acc1, false, false);
    acc2 = __builtin_amdgcn_wmma_f32_16x16x4_f32(false, a, false, b2, (short)0, acc2, false, false);
    acc3 = __builtin_amdgcn_wmma_f32_16x16x4_f32(false, a, false, b3, (short)0, acc3, false, false);
  }

  v8f accs[4] = {acc0, acc1, acc2, acc3};
  #pragma unroll
  for (int tI = 0; tI < 4; ++tI) {
    const int g = n0 + tI * 16 + l16;
    const float bias = bih[g] + bhh[g];
    #pragma unroll
    for (int r = 0; r < 8; ++r) {
      const int row = m0 + r + 8 * half;
      xg[(size_t)row * G4 + g] = accs[tI][r] + bias;
    }
  }
}

// ---------------------------------------------------------------------------
// Phase 2+3: sequential LSTM scan. One workgroup, 512 threads (16 waves).
// Thread t owns gate row t of W_hh: 128 fp32 weights held in VGPRs.
// h[128] lives in LDS (broadcast reads). Matvec uses v_pk_fma_f32 (2 MAC/clk).
// x_gates row streamed with a 4-step register prefetch pipeline.
// ---------------------------------------------------------------------------
__device__ __forceinline__ float fsigmoid(float v) {
  v = fminf(fmaxf(v, -30.f), 30.f);
  return 1.f / (1.f + __expf(-v));
}
__device__ __forceinline__ float ftanh(float v) {
  v = fminf(fmaxf(v, -15.f), 15.f);
  float e = __expf(-2.f * v);
  return (1.f - e) / (1.f + e);
}

__global__ __launch_bounds__(512) void lstm_scan(
    const float* __restrict__ xg,    // [SEQ, G4]
    const float* __restrict__ Whh,   // [G4, HD]
    const float* __restrict__ Wlin,  // [HD]
    const float* __restrict__ blin,  // [1]
    float* __restrict__ out)         // [1]
{
  __shared__ float hbuf[HD];
  __shared__ float act[G4];
  const int t = threadIdx.x;

  // Resident weights: row t of W_hh as 64 packed float2 (128 VGPRs).
  v2f w2[HD / 2];
  #pragma unroll
  for (int i = 0; i < HD / 2; ++i)
    w2[i] = ((const v2f*)(Whh + (size_t)t * HD))[i];

  float c = 0.f;
  if (t < HD) hbuf[t] = 0.f;
  __syncthreads();

  const v2f* h2 = (const v2f*)hbuf;

  auto step = [&](float xval) {
    v2f accA = {0.f, 0.f};
    v2f accB = {0.f, 0.f};
    #pragma unroll
    for (int kk = 0; kk < HD / 2; kk += 2) {
      v2f hv0 = h2[kk];
      v2f hv1 = h2[kk + 1];
      // VOP3P packed fp32 FMA: two MACs per instruction per lane.
      asm("v_pk_fma_f32 %0, %1, %2, %0" : "+v"(accA) : "v"(w2[kk]),     "v"(hv0));
      asm("v_pk_fma_f32 %0, %1, %2, %0" : "+v"(accB) : "v"(w2[kk + 1]), "v"(hv1));
    }
    float a = accA.x + accA.y + accB.x + accB.y + xval;
    // gates: [0,128)=i sig, [128,256)=f sig, [256,384)=g tanh, [384,512)=o sig
    float av = (t >= 2 * HD && t < 3 * HD) ? ftanh(a) : fsigmoid(a);
    act[t] = av;
    __syncthreads();
    if (t < HD) {
      float gi = act[t];
      float gf = act[t + HD];
      float gg = act[t + 2 * HD];
      float go = act[t + 3 * HD];
      c = gf * c + gi * gg;
      hbuf[t] = go * ftanh(c);
    }
    __syncthreads();
  };

  // 4-deep register prefetch pipeline over x_gates rows.
  float x0 = xg[(size_t)0 * G4 + t];
  float x1 = xg[(size_t)1 * G4 + t];
  float x2 = xg[(size_t)2 * G4 + t];
  float x3 = xg[(size_t)3 * G4 + t];

  for (int s = 0; s < SEQ; s += 4) {
    step(x0); x0 = (s + 4 < SEQ) ? xg[(size_t)(s + 4) * G4 + t] : 0.f;
    step(x1); x1 = (s + 5 < SEQ) ? xg[(size_t)(s + 5) * G4 + t] : 0.f;
    step(x2); x2 = (s + 6 < SEQ) ? xg[(size_t)(s + 6) * G4 + t] : 0.f;
    step(x3); x3 = (s + 7 < SEQ) ? xg[(size_t)(s + 7) * G4 + t] : 0.f;
  }

  // Phase 3: final linear + sigmoid on h_T (in hbuf).
  if (t == 0) {
    float d = blin[0];
    #pragma unroll 8
    for (int j = 0; j < HD; ++j) d += hbuf[j] * Wlin[j];
    out[0] = 1.f / (1.f + __expf(-d));
  }
}

// ---------------------------------------------------------------------------
extern "C" void kernel_launch(void* const* d_in, const int* in_sizes, int n_in,
                              void* d_out, int out_size, void* d_ws, size_t ws_size,
                              hipStream_t stream) {
  const float* x    = (const float*)d_in[0];  // [SEQ, IN]
  const float* Wih  = (const float*)d_in[1];  // [4H, IN]
  const float* Whh  = (const float*)d_in[2];  // [4H, H]
  const float* bih  = (const float*)d_in[3];  // [4H]
  const float* bhh  = (const float*)d_in[4];  // [4H]
  const float* Wlin = (const float*)d_in[5];  // [1, H]
  const float* blin = (const float*)d_in[6];  // [1]

  float* xg = (float*)d_ws;                   // needs SEQ*G4*4 = 128 MB scratch

  xgates_gemm<<<SEQ / 16, 256, 0, stream>>>(x, Wih, bih, bhh, xg);
  lstm_scan<<<1, 512, 0, stream>>>(xg, Whh, Wlin, blin, (float*)d_out);
}